// PairwiseGeometryModule_85667417686315
// MI455X (gfx1250) — compile-verified
//
#include <hip/hip_runtime.h>
#include <math.h>

typedef __attribute__((ext_vector_type(2))) float v2f;
typedef __attribute__((ext_vector_type(8))) float v8f;

// ---- output layout (flat concat, fp32) ----
#define OFF_DELTA 0          // (2,512,512,3) = 1572864
#define OFF_DIST  1572864    // (2,512,512)
#define OFF_DIST2 2097152
#define OFF_NORM  2621440
#define OFF_LAT   3145728    // (2,6)
#define OFF_GRAM  3145740    // (2,3,3)
#define OFF_G6    3145758    // (2,6)

__device__ __forceinline__ float nz(float x) { return isfinite(x) ? x : 0.0f; }

// product(range(-1,2), repeat=3): k slowest->fastest = (x,y,z)
__device__ __forceinline__ float offx(int k) { return (float)(k / 9 - 1); }
__device__ __forceinline__ float offy(int k) { return (float)((k / 3) % 3 - 1); }
__device__ __forceinline__ float offz(int k) { return (float)(k % 3 - 1); }

struct GramT { float g00, g11, g22, g01, g02, g12, scale, l0, l1, l2, l3, l4, l5; };

__device__ __forceinline__ GramT make_gram(const float* __restrict__ lattice, int b) {
    GramT G;
    const float* L = lattice + b * 6;
    G.l0 = fminf(fmaxf(L[0], -5.0f), 5.0f);
    G.l1 = fminf(fmaxf(L[1], -5.0f), 5.0f);
    G.l2 = fminf(fmaxf(L[2], -5.0f), 5.0f);
    G.l3 = fminf(fmaxf(L[3], -0.9999f), 0.9999f);
    G.l4 = fminf(fmaxf(L[4], -0.9999f), 0.9999f);
    G.l5 = fminf(fmaxf(L[5], -0.9999f), 0.9999f);
    float a = __expf(G.l0), bb = __expf(G.l1), c = __expf(G.l2);
    G.g00 = nz(a * a);  G.g11 = nz(bb * bb);  G.g22 = nz(c * c);
    G.g01 = nz(a * bb * G.l5);   // a*b*cos(gamma)
    G.g02 = nz(a * c * G.l4);    // a*c*cos(beta)
    G.g12 = nz(bb * c * G.l3);   // b*c*cos(alpha)
    G.scale = fmaxf((a + bb + c) * (1.0f / 3.0f), 1e-6f);
    return G;
}

__device__ __forceinline__ float cof(int k, const GramT& G) {
    float ox = offx(k), oy = offy(k), oz = offz(k);
    return G.g00 * ox * ox + G.g11 * oy * oy + G.g22 * oz * oz +
           2.0f * (G.g01 * ox * oy + G.g02 * ox * oz + G.g12 * oy * oz);
}

// One wave = fixed (b,n); loops 8 m-tiles of 16 pairs. Per tile, the 27
// periodic-image distances are two V_WMMA_F32_16X16X4_F32 ops:
//   D(27x16 pairs) = OFF(27x4) * FEAT(4x16) + c_k
// A/C/gram/coords[n] setup is hoisted out of the tile loop.
__global__ __launch_bounds__(256)
void pair_geom_kernel(const float* __restrict__ coords,
                      const float* __restrict__ lattice,
                      float* __restrict__ out) {
    const int lane = threadIdx.x & 31;
    const int gw   = blockIdx.x * (blockDim.x >> 5) + (threadIdx.x >> 5); // 0..4095
    const int b    = gw >> 11;            // 2048 waves per batch
    const int rem  = gw & 2047;
    const int n    = rem >> 2;
    const int q    = rem & 3;             // quarter of the m-row
    const bool hi  = lane >= 16;
    const int lm   = lane & 15;

    const GramT G = make_gram(lattice, b);

    // ---- hoisted setup -------------------------------------------------
    // A tiles (16x4, M x K): row M = image index; VGPR0 = K0/K2, VGPR1 = K1/K3
    v2f A1, A2;
    A1.x = hi ? offz(lm) : offx(lm);
    A1.y = hi ? 1.0f     : offy(lm);
    const int k2 = 16 + lm;
    if (k2 < 27) { A2.x = hi ? offz(k2) : offx(k2); A2.y = hi ? 1.0f : offy(k2); }
    else         { A2.x = 0.0f; A2.y = 0.0f; }

    // C tiles: per-row constant c_k = o^T G o ; pad rows -> +1e30
    v8f C1, C2;
#pragma unroll
    for (int v = 0; v < 8; ++v) {
        int i1 = v + (hi ? 8 : 0);
        int i2 = 16 + v + (hi ? 8 : 0);
        C1[v] = cof(i1, G);
        C2[v] = (i2 < 27) ? cof(i2, G) : 1e30f;
    }

    const float* cn = coords + ((b << 9) + n) * 3;
    const float cnx = cn[0], cny = cn[1], cnz = cn[2];

    const long long Pn = ((long long)(b << 9) + n) * 512;

    // ---- tile loop: 8 tiles x 16 pairs --------------------------------
    for (int t = 0; t < 8; ++t) {
        const int m = (q << 7) + (t << 4) + lm;
        const float* cm = coords + ((b << 9) + m) * 3;
        const float dx = cnx - cm[0];
        const float dy = cny - cm[1];
        const float dz = cnz - cm[2];

        // features f = [2*(G d)x, 2*(G d)y, 2*(G d)z, d.(G d)]
        const float gx = G.g00 * dx + G.g01 * dy + G.g02 * dz;
        const float gy = G.g01 * dx + G.g11 * dy + G.g12 * dz;
        const float gz = G.g02 * dx + G.g12 * dy + G.g22 * dz;
        const float base = dx * gx + dy * gy + dz * gz;

        // B (4x16, K x N): VGPR0 = K0 (lo lanes) / K2 (hi lanes); VGPR1 = K1 / K3
        v2f Bm;
        Bm.x = hi ? 2.0f * gz : 2.0f * gx;
        Bm.y = hi ? base      : 2.0f * gy;

        v8f D1 = __builtin_amdgcn_wmma_f32_16x16x4_f32(false, A1, false, Bm, (short)0, C1, false, false);
        v8f D2 = __builtin_amdgcn_wmma_f32_16x16x4_f32(false, A2, false, Bm, (short)0, C2, false, false);

        // per-lane argmin (indices visited in increasing order -> first-min on ties)
        float bestv = D1[0];
        int   besti = hi ? 8 : 0;
#pragma unroll
        for (int v = 1; v < 8; ++v) {
            float val = D1[v];
            int   idx = v + (hi ? 8 : 0);
            if (val < bestv) { bestv = val; besti = idx; }
        }
#pragma unroll
        for (int v = 0; v < 8; ++v) {
            float val = D2[v];
            int   idx = 16 + v + (hi ? 8 : 0);
            if (val < bestv) { bestv = val; besti = idx; }
        }

        // merge row-halves: lane l <-> lane l^16; prefer smaller index on tie
        float ov = __shfl_xor(bestv, 16);
        int   oi = __shfl_xor(besti, 16);
        if (ov < bestv || (ov == bestv && oi < besti)) { bestv = ov; besti = oi; }

        if (!hi) {
            const long long P = Pn + m;
            float d2 = nz(fmaxf(bestv, 0.0f));
            float d1 = nz(sqrtf(d2));
            float dn = nz(d1 / G.scale);
            out[OFF_DELTA + P * 3 + 0] = dx + offx(besti);
            out[OFF_DELTA + P * 3 + 1] = dy + offy(besti);
            out[OFF_DELTA + P * 3 + 2] = dz + offz(besti);
            out[OFF_DIST  + P] = d1;
            out[OFF_DIST2 + P] = d2;
            out[OFF_NORM  + P] = dn;
        }
    }
}

// tiny per-batch outputs: lattice_y1, gram (3x3 row-major), gram6
__global__ void pair_geom_small(const float* __restrict__ lattice, float* __restrict__ out) {
    int b = threadIdx.x;
    if (b >= 2) return;
    GramT G = make_gram(lattice, b);
    float ly[6] = { nz(G.l0), nz(G.l1), nz(G.l2), nz(G.l3), nz(G.l4), nz(G.l5) };
#pragma unroll
    for (int i = 0; i < 6; ++i) out[OFF_LAT + b * 6 + i] = ly[i];
    float gm[9] = { G.g00, G.g01, G.g02,
                    G.g01, G.g11, G.g12,
                    G.g02, G.g12, G.g22 };
#pragma unroll
    for (int i = 0; i < 9; ++i) out[OFF_GRAM + b * 9 + i] = gm[i];
    float g6[6] = { G.g00, G.g11, G.g22, G.g01, G.g02, G.g12 };
#pragma unroll
    for (int i = 0; i < 6; ++i) out[OFF_G6 + b * 6 + i] = g6[i];
}

extern "C" void kernel_launch(void* const* d_in, const int* in_sizes, int n_in,
                              void* d_out, int out_size, void* d_ws, size_t ws_size,
                              hipStream_t stream) {
    const float* coords  = (const float*)d_in[0];   // (2,512,3)
    const float* lattice = (const float*)d_in[1];   // (2,6)
    float* out = (float*)d_out;
    // 4096 waves total (each does 128 pairs), 8 waves (256 threads) per block
    pair_geom_kernel<<<512, 256, 0, stream>>>(coords, lattice, out);
    pair_geom_small<<<1, 64, 0, stream>>>(lattice, out);
    (void)in_sizes; (void)n_in; (void)out_size; (void)d_ws; (void)ws_size;
}